// GCN_23124103922098
// MI455X (gfx1250) — compile-verified
//
#include <hip/hip_runtime.h>
#include <hip/hip_bf16.h>

// ---------------------------------------------------------------------------
// GCN (3x GCNConv) for MI455X / gfx1250.
//   - GEMMs via V_WMMA_F32_16X16X4_F32, two independent accumulator chains
//   - W staged in LDS (padded to 16 cols so F_out=8 reuses the 16-wide tile)
//   - per-edge norm coefficients precomputed once, reused by all 3 layers
//   - edge scatter-add via global_atomic_add_f32 (all buffers fit in 192MB L2)
// ---------------------------------------------------------------------------

typedef __attribute__((ext_vector_type(2))) float v2f;
typedef __attribute__((ext_vector_type(8))) float v8f;

#ifndef GCN_F_IN_MAX
#define GCN_F_IN_MAX 256
#endif

// ---- degree / normalization ----------------------------------------------

__global__ void k_init_deg(float* __restrict__ deg, int n) {
    int i = blockIdx.x * blockDim.x + threadIdx.x;
    if (i < n) deg[i] = 1.0f;  // self-loop
}

__global__ void k_deg_scatter(const int* __restrict__ ei, float* __restrict__ deg, int E) {
    int i = blockIdx.x * blockDim.x + threadIdx.x;
    if (i < E) atomicAdd(&deg[ei[E + i]], 1.0f);  // dst row of edge_index
}

// deg -> dinv = rsqrt(deg), selfc = 1/deg  (selfc aliases the deg buffer)
__global__ void k_finalize_norm(float* __restrict__ dinv, float* __restrict__ deg_selfc, int n) {
    int i = blockIdx.x * blockDim.x + threadIdx.x;
    if (i < n) {
        float d = deg_selfc[i];
        dinv[i] = rsqrtf(d);
        deg_selfc[i] = 1.0f / d;
    }
}

// norm_e[e] = dinv[src[e]] * dinv[dst[e]]  (computed once, reused 3x)
__global__ void k_edge_norm(const int* __restrict__ ei, const float* __restrict__ dinv,
                            float* __restrict__ ne, int E) {
    int i = blockIdx.x * blockDim.x + threadIdx.x;
    if (i < E) ne[i] = dinv[ei[i]] * dinv[ei[E + i]];
}

__global__ void k_zero(float* __restrict__ p, long n) {
    long i = (long)blockIdx.x * blockDim.x + threadIdx.x;
    if (i < n) p[i] = 0.0f;
}

// ---- WMMA GEMM: out[N x 16(stride)] = X[N x K] @ Wpad[K x 16] -------------
// One wave computes one 16x16 output tile. K in {256, 16}. Cols >= Fout are 0.
// Two accumulator chains break the serial WMMA RAW dependency.

__global__ __launch_bounds__(256) void k_gemm_wmma(const float* __restrict__ X,
                                                   const float* __restrict__ W,
                                                   float* __restrict__ out,
                                                   int Nn, int nTiles, int K, int Fout) {
    __shared__ float sW[GCN_F_IN_MAX * 16];

    const int tid  = threadIdx.x;
    const int lane = tid & 31;
    const int wave = tid >> 5;

    // Stage W into LDS, zero-padded to 16 columns.
    for (int idx = tid; idx < K * 16; idx += blockDim.x) {
        int kk = idx >> 4, nn = idx & 15;
        sW[idx] = (nn < Fout) ? W[kk * Fout + nn] : 0.0f;
    }
    __syncthreads();

    const int tile = blockIdx.x * (blockDim.x >> 5) + wave;
    if (tile >= nTiles) return;  // wave-uniform: EXEC stays all-1s for WMMA

    const int mrow = tile * 16 + (lane & 15);          // A row (M = lane%16)
    const int row  = mrow < Nn ? mrow : (Nn - 1);      // safety clamp (loads only)
    const int klo  = (lane >> 4) << 1;                 // K sub-offset: 0 or 2
    const int n    = lane & 15;                        // B/D column

    const float* __restrict__ xrow = X + (long)row * K;

    v8f acc0 = {}, acc1 = {};
    int k = 0;
    for (; k + 8 <= K; k += 8) {
        v2f a0, b0, a1, b1;
        a0.x = xrow[k + klo];
        a0.y = xrow[k + klo + 1];
        b0.x = sW[(k + klo) * 16 + n];
        b0.y = sW[(k + klo + 1) * 16 + n];
        a1.x = xrow[k + 4 + klo];
        a1.y = xrow[k + 4 + klo + 1];
        b1.x = sW[(k + 4 + klo) * 16 + n];
        b1.y = sW[(k + 4 + klo + 1) * 16 + n];
        // speculative WGP-scope prefetch eight loop iterations (64 floats) ahead
        __builtin_prefetch(&xrow[k + klo + 64], 0, 3);
        acc0 = __builtin_amdgcn_wmma_f32_16x16x4_f32(
            false, a0, false, b0, (short)0, acc0, false, false);
        acc1 = __builtin_amdgcn_wmma_f32_16x16x4_f32(
            false, a1, false, b1, (short)0, acc1, false, false);
    }
    for (; k < K; k += 4) {  // tail (unused for K in {16,256})
        v2f a, b;
        a.x = xrow[k + klo];
        a.y = xrow[k + klo + 1];
        b.x = sW[(k + klo) * 16 + n];
        b.y = sW[(k + klo + 1) * 16 + n];
        acc0 = __builtin_amdgcn_wmma_f32_16x16x4_f32(
            false, a, false, b, (short)0, acc0, false, false);
    }

    // D layout: VGPR r, lane l -> M = r + 8*(l>>4), N = l&15
    const int h = lane >> 4;
    const int mbase = tile * 16 + 8 * h;
#pragma unroll
    for (int r = 0; r < 8; ++r) {
        int m = mbase + r;
        if (m < Nn) out[(long)m * 16 + n] = acc0[r] + acc1[r];
    }
}

// ---- edge scatter: agg[dst] += xw[src] * norm_e[e] ------------------------
// One feature per lane; lf = log2(F) in {4,3}. Buffers have stride 16.

__global__ void k_edge_scatter(const int* __restrict__ ei,
                               const float* __restrict__ ne,
                               const float* __restrict__ xw,
                               float* __restrict__ agg,
                               int E, int lf) {
    long t = (long)blockIdx.x * blockDim.x + threadIdx.x;
    long total = (long)E << lf;
    if (t >= total) return;
    int e = (int)(t >> lf);
    int f = (int)(t & ((1 << lf) - 1));
    int s = ei[e];        // broadcast across the 2^lf lanes of this edge
    int d = ei[E + e];
    float w = ne[e];
    atomicAdd(&agg[(long)d * 16 + f], xw[(long)s * 16 + f] * w);
}

// ---- combine: h = [relu](agg + selfc*xw + b), stride-16 buffers -----------

__global__ void k_combine(const float* __restrict__ agg,
                          const float* __restrict__ xw,
                          const float* __restrict__ selfc,
                          const float* __restrict__ bias,
                          float* __restrict__ hout,
                          int Nn, int lf, int do_relu) {
    long t = (long)blockIdx.x * blockDim.x + threadIdx.x;
    long total = (long)Nn << lf;
    if (t >= total) return;
    int node = (int)(t >> lf);
    int f = (int)(t & ((1 << lf) - 1));
    long idx = (long)node * 16 + f;
    float v = agg[idx] + selfc[node] * xw[idx] + bias[f];
    if (do_relu) v = fmaxf(v, 0.0f);
    hout[idx] = v;
}

// ---- log_softmax over 8 classes (input stride 16, output stride 8) --------

__global__ void k_logsoftmax8(const float* __restrict__ h, float* __restrict__ out, int Nn) {
    int i = blockIdx.x * blockDim.x + threadIdx.x;
    if (i >= Nn) return;
    const float* r = h + (long)i * 16;
    float m = r[0];
#pragma unroll
    for (int c = 1; c < 8; ++c) m = fmaxf(m, r[c]);
    float s = 0.0f;
#pragma unroll
    for (int c = 0; c < 8; ++c) s += expf(r[c] - m);
    float ls = logf(s);
#pragma unroll
    for (int c = 0; c < 8; ++c) out[(long)i * 8 + c] = r[c] - m - ls;
}

// ---------------------------------------------------------------------------

static inline int ceil_div(long a, int b) { return (int)((a + b - 1) / b); }

extern "C" void kernel_launch(void* const* d_in, const int* in_sizes, int n_in,
                              void* d_out, int out_size, void* d_ws, size_t ws_size,
                              hipStream_t stream) {
    const float* x  = (const float*)d_in[0];
    const int*   ei = (const int*)d_in[1];   // [2, E] int32 (JAX x64 disabled)
    const float* W1 = (const float*)d_in[2];
    const float* b1 = (const float*)d_in[3];
    const float* W2 = (const float*)d_in[4];
    const float* b2 = (const float*)d_in[5];
    const float* W3 = (const float*)d_in[6];
    const float* b3 = (const float*)d_in[7];
    float* out = (float*)d_out;

    const int Nn = in_sizes[0] / 256;   // 100000
    const int E  = in_sizes[1] / 2;     // 3200000
    const int K1 = 256, H = 16, C = 8;

    // workspace (floats): dinv[N] | selfc[N] | ne[E] | xw[16N] | agg[16N] | h[16N]
    float* ws    = (float*)d_ws;
    float* dinv  = ws;
    float* selfc = ws + Nn;
    float* ne    = selfc + Nn;
    float* xw    = ne + E;
    float* agg   = xw + 16L * Nn;
    float* hbuf  = agg + 16L * Nn;

    const int B = 256;
    const int nTiles = (Nn + 15) / 16;
    const int gemmBlocks = ceil_div(nTiles, B / 32);

    // --- normalization coefficients (once, reused by all layers) ---
    k_init_deg<<<ceil_div(Nn, B), B, 0, stream>>>(selfc, Nn);
    k_deg_scatter<<<ceil_div(E, B), B, 0, stream>>>(ei, selfc, E);
    k_finalize_norm<<<ceil_div(Nn, B), B, 0, stream>>>(dinv, selfc, Nn);
    k_edge_norm<<<ceil_div(E, B), B, 0, stream>>>(ei, dinv, ne, E);

    // --- layer 1: K=256 -> 16, ReLU ---
    k_gemm_wmma<<<gemmBlocks, B, 0, stream>>>(x, W1, xw, Nn, nTiles, K1, H);
    k_zero<<<ceil_div(16L * Nn, B), B, 0, stream>>>(agg, 16L * Nn);
    k_edge_scatter<<<ceil_div((long)E * 16, B), B, 0, stream>>>(ei, ne, xw, agg, E, 4);
    k_combine<<<ceil_div((long)Nn * 16, B), B, 0, stream>>>(agg, xw, selfc, b1, hbuf, Nn, 4, 1);

    // --- layer 2: K=16 -> 16, ReLU ---
    k_gemm_wmma<<<gemmBlocks, B, 0, stream>>>(hbuf, W2, xw, Nn, nTiles, H, H);
    k_zero<<<ceil_div(16L * Nn, B), B, 0, stream>>>(agg, 16L * Nn);
    k_edge_scatter<<<ceil_div((long)E * 16, B), B, 0, stream>>>(ei, ne, xw, agg, E, 4);
    k_combine<<<ceil_div((long)Nn * 16, B), B, 0, stream>>>(agg, xw, selfc, b2, hbuf, Nn, 4, 1);

    // --- layer 3: K=16 -> 8, no ReLU (cols 8..15 of W zero-padded in LDS) ---
    k_gemm_wmma<<<gemmBlocks, B, 0, stream>>>(hbuf, W3, xw, Nn, nTiles, H, C);
    k_zero<<<ceil_div(16L * Nn, B), B, 0, stream>>>(agg, 16L * Nn);
    k_edge_scatter<<<ceil_div((long)E * 8, B), B, 0, stream>>>(ei, ne, xw, agg, E, 3);
    k_combine<<<ceil_div((long)Nn * 8, B), B, 0, stream>>>(agg, xw, selfc, b3, hbuf, Nn, 3, 0);

    // --- log_softmax over 8 classes ---
    k_logsoftmax8<<<ceil_div(Nn, B), B, 0, stream>>>(hbuf, out, Nn);
}